// CombinedModel_90460601188989
// MI455X (gfx1250) — compile-verified
//
#include <hip/hip_runtime.h>

// ---------------------------------------------------------------------------
// RNODE (3-dim Euler recurrence, 500k steps) + per-step 3->64->3 MLP readout.
//
// MI455X reasoning:
//  * MLP part: 0.38 GFLOP / ~40MB traffic -> ~2us at 23.3 TB/s. Irrelevant.
//  * Recurrence: nonlinear (sigmoid) -> no parallel scan. 500k dependent
//    steps on one wave is the latency floor; minimize the per-step chain
//    (fma -> v_exp_f32 -> add -> v_rcp_f32 -> fma -> add).
//  * Readout uses V_WMMA_F32_16X16X4_F32 (exact f32). K padded 3->4, and the
//    pad slot carries the fc1 bias (A[:,3]=b1, B[3,:]=1) so the accumulator
//    is the inline constant 0 (no Cinit registers, no spills).
// ---------------------------------------------------------------------------

typedef __attribute__((ext_vector_type(2))) float v2f;
typedef __attribute__((ext_vector_type(8))) float v8f;

#define RN_DT 0.01f
#define NL2E (-1.44269504088896340736f)   // -log2(e)

// ---------------------------------------------------------------------------
// Kernel 1: dv[c][t] = DT * (in_mat @ u[:,t])   (parallel, memory bound)
// ---------------------------------------------------------------------------
__global__ void rnode_drive_kernel(const float* __restrict__ u,
                                   const float* __restrict__ in_mat,
                                   float* __restrict__ dv, int T) {
  int t = blockIdx.x * blockDim.x + threadIdx.x;
  int stride = gridDim.x * blockDim.x;
  const float m00 = in_mat[0], m01 = in_mat[1], m02 = in_mat[2];
  const float m10 = in_mat[3], m11 = in_mat[4], m12 = in_mat[5];
  const float m20 = in_mat[6], m21 = in_mat[7], m22 = in_mat[8];
  for (; t < T; t += stride) {
    float u0 = u[t], u1 = u[T + t], u2 = u[2 * T + t];
    dv[t]         = RN_DT * (m00 * u0 + m01 * u1 + m02 * u2);
    dv[T + t]     = RN_DT * (m10 * u0 + m11 * u1 + m12 * u2);
    dv[2 * T + t] = RN_DT * (m20 * u0 + m21 * u1 + m22 * u2);
  }
}

// ---------------------------------------------------------------------------
// Kernel 2: sequential Euler scan, one wave, all lanes redundant.
// Lanes 0..2 stream the potentials (async stores, off the chain).
// USE_DV: read precomputed drive; else fold in_mat@u into the loop.
// ---------------------------------------------------------------------------
template <bool USE_DV>
__global__ void rnode_scan_kernel(const float* __restrict__ src,   // dv or u
                                  const float* __restrict__ in_mat,
                                  const float* __restrict__ W,
                                  const float* __restrict__ theta,
                                  const float* __restrict__ tau,
                                  float* __restrict__ pot, int T) {
  const int lane = threadIdx.x & 31;
  const float a = 1.0f - RN_DT / tau[0];                 // decay factor
  const float w00 = RN_DT * W[0], w01 = RN_DT * W[1], w02 = RN_DT * W[2];
  const float w10 = RN_DT * W[3], w11 = RN_DT * W[4], w12 = RN_DT * W[5];
  const float w20 = RN_DT * W[6], w21 = RN_DT * W[7], w22 = RN_DT * W[8];
  const float c0 = theta[0] * NL2E, c1 = theta[1] * NL2E, c2 = theta[2] * NL2E;
  float m00 = 0.f, m01 = 0.f, m02 = 0.f;
  float m10 = 0.f, m11 = 0.f, m12 = 0.f;
  float m20 = 0.f, m21 = 0.f, m22 = 0.f;
  if (!USE_DV) {
    m00 = RN_DT * in_mat[0]; m01 = RN_DT * in_mat[1]; m02 = RN_DT * in_mat[2];
    m10 = RN_DT * in_mat[3]; m11 = RN_DT * in_mat[4]; m12 = RN_DT * in_mat[5];
    m20 = RN_DT * in_mat[6]; m21 = RN_DT * in_mat[7]; m22 = RN_DT * in_mat[8];
  }

  float x0 = 0.f, x1 = 0.f, x2 = 0.f;
#pragma unroll 4
  for (int t = 0; t < T; ++t) {
    float d0, d1, d2;
    if (USE_DV) {
      d0 = src[t]; d1 = src[T + t]; d2 = src[2 * T + t];
    } else {
      float u0 = src[t], u1 = src[T + t], u2 = src[2 * T + t];
      d0 = m00 * u0 + m01 * u1 + m02 * u2;
      d1 = m10 * u0 + m11 * u1 + m12 * u2;
      d2 = m20 * u0 + m21 * u1 + m22 * u2;
    }
    // sigmoid(x + theta) = rcp(1 + exp2(-(x+theta)*log2e))  (3 independent
    // chains, pipelined; this is the per-step latency floor)
    float s0 = __builtin_amdgcn_rcpf(1.0f + __builtin_amdgcn_exp2f(fmaf(x0, NL2E, c0)));
    float s1 = __builtin_amdgcn_rcpf(1.0f + __builtin_amdgcn_exp2f(fmaf(x1, NL2E, c1)));
    float s2 = __builtin_amdgcn_rcpf(1.0f + __builtin_amdgcn_exp2f(fmaf(x2, NL2E, c2)));
    // balanced tree: only fma+add after the last sigmoid is ready
    float b0 = fmaf(a, x0, d0), b1 = fmaf(a, x1, d1), b2 = fmaf(a, x2, d2);
    float q0 = fmaf(w01, s1, w02 * s2);
    float q1 = fmaf(w11, s1, w12 * s2);
    float q2 = fmaf(w21, s1, w22 * s2);
    x0 = fmaf(w00, s0, b0) + q0;
    x1 = fmaf(w10, s0, b1) + q1;
    x2 = fmaf(w20, s0, b2) + q2;
    float v = (lane == 0) ? x0 : ((lane == 1) ? x1 : x2);
    if (lane < 3) pot[lane * T + t] = v;
  }
}

// ---------------------------------------------------------------------------
// Kernel 3: readout out[:,t] = fc2 @ relu(fc1 @ x_t + b1) + b2 for all t,
// using V_WMMA_F32_16X16X4_F32 for fc1 (16 timesteps per tile).
// Layout assumptions per CDNA5 ISA 7.12.2:
//   A 16x4: lanes 0-15 row M, vgpr{0,1}={K0,K1}; lanes 16-31 vgpr{0,1}={K2,K3}
//   B 4x16: lanes 0-15 col N, vgpr{0,1}={K0,K1}; lanes 16-31 vgpr{0,1}={K2,K3}
//   C 16x16: vgpr r holds M=r (lanes 0-15) / M=r+8 (lanes 16-31), N=lane&15
// K slot 3 carries the bias: A[:,3]=fc1_b, B[3,:]=1.0  => C starts at 0.
// fc2 (3 rows only) is VALU dot products on the C-layout registers, combining
// the M / M+8 lane-halves with one __shfl_xor(16).
// ---------------------------------------------------------------------------
__global__ void rnode_readout_kernel(const float* __restrict__ pot,
                                     const float* __restrict__ fc1_w,
                                     const float* __restrict__ fc1_b,
                                     const float* __restrict__ fc2_w,
                                     const float* __restrict__ fc2_b,
                                     float* __restrict__ out, int T) {
  const int lane = threadIdx.x & 31;
  const int n = lane & 15;
  const int hi = lane >> 4;
  const int wavesPerBlock = blockDim.x >> 5;
  const int wave = blockIdx.x * wavesPerBlock + (threadIdx.x >> 5);
  const int nwaves = gridDim.x * wavesPerBlock;
  const int ntiles = (T + 15) >> 4;
  const float INFP = __builtin_inff();

  // fc1 A tiles (64x3 padded to 64x4 with bias in K=3), loop-invariant.
  // Branchless: select the column indices per lane-half.
  v2f A[4];
#pragma unroll
  for (int mt = 0; mt < 4; ++mt) {
    int R = mt * 16 + n;
    float ax0 = fc1_w[R * 3 + 0], ay0 = fc1_w[R * 3 + 1];   // K0,K1 (lo half)
    float ax1 = fc1_w[R * 3 + 2], ay1 = fc1_b[R];           // K2,K3 (hi half)
    A[mt].x = hi ? ax1 : ax0;
    A[mt].y = hi ? ay1 : ay0;
  }

  const float b20 = fc2_b[0], b21 = fc2_b[1], b22 = fc2_b[2];
  const int planeX = hi ? 2 * T : 0;   // B.x source plane (x0 or x2)

  for (int tile = wave; tile < ntiles; tile += nwaves) {
    int t = tile * 16 + n;
    int tc = t < T ? t : T - 1;      // clamp: tail columns computed, not stored
    // B tile, branchless: lo half = {x0[t], x1[t]}, hi half = {x2[t], 1.0}.
    v2f B;
    B.x = pot[planeX + tc];
    float by = pot[T + tc];
    B.y = hi ? 1.0f : by;

    float p0 = 0.0f, p1 = 0.0f, p2 = 0.0f;
#pragma unroll
    for (int mt = 0; mt < 4; ++mt) {
      v8f c = {};   // bias already inside the WMMA via the K=3 slot
      c = __builtin_amdgcn_wmma_f32_16x16x4_f32(false, A[mt], false, B,
                                                (short)0, c, false, false);
#pragma unroll
      for (int r = 0; r < 8; ++r) {
        float h = __builtin_amdgcn_fmed3f(c[r], 0.0f, INFP);  // relu, 1 op
        int M = mt * 16 + 8 * hi + r;              // hidden index of this elem
        p0 = fmaf(fc2_w[M], h, p0);
        p1 = fmaf(fc2_w[64 + M], h, p1);
        p2 = fmaf(fc2_w[128 + M], h, p2);
      }
    }
    // combine the M / M+8 lane-halves (same timestep column n).
    p0 += __shfl_xor(p0, 16);
    p1 += __shfl_xor(p1, 16);
    p2 += __shfl_xor(p2, 16);
    if (hi == 0 && t < T) {
      out[t]         = p0 + b20;
      out[T + t]     = p1 + b21;
      out[2 * T + t] = p2 + b22;
    }
  }
}

// ---------------------------------------------------------------------------
// Host launcher.  d_out = [outputs(3*T) | potentials(3*T)] f32.
// d_ws holds dv (3*T f32) when large enough; otherwise the scan folds the
// drive term in directly.
// ---------------------------------------------------------------------------
extern "C" void kernel_launch(void* const* d_in, const int* in_sizes, int n_in,
                              void* d_out, int out_size, void* d_ws, size_t ws_size,
                              hipStream_t stream) {
  const float* u      = (const float*)d_in[0];
  const float* in_mat = (const float*)d_in[1];
  const float* W      = (const float*)d_in[2];
  const float* theta  = (const float*)d_in[3];
  const float* tau    = (const float*)d_in[4];
  const float* fc1_w  = (const float*)d_in[5];
  const float* fc1_b  = (const float*)d_in[6];
  const float* fc2_w  = (const float*)d_in[7];
  const float* fc2_b  = (const float*)d_in[8];
  float* out = (float*)d_out;

  const int T = in_sizes[0] / 3;           // 500000
  float* pot = out + (size_t)3 * T;        // potentials half of d_out
  float* dv  = (float*)d_ws;

  const bool use_dv = (ws_size >= (size_t)3 * T * sizeof(float));
  if (use_dv) {
    int grid = (T + 255) / 256;
    if (grid > 2048) grid = 2048;
    rnode_drive_kernel<<<grid, 256, 0, stream>>>(u, in_mat, dv, T);
    rnode_scan_kernel<true><<<1, 32, 0, stream>>>(dv, in_mat, W, theta, tau, pot, T);
  } else {
    rnode_scan_kernel<false><<<1, 32, 0, stream>>>(u, in_mat, W, theta, tau, pot, T);
  }
  rnode_readout_kernel<<<256, 256, 0, stream>>>(pot, fc1_w, fc1_b, fc2_w, fc2_b, out, T);
}